// Linear_45681272160662
// MI455X (gfx1250) — compile-verified
//
#include <hip/hip_runtime.h>
#include <stdint.h>

#define B_   128
#define K_   2048
#define D_   10
#define TPB  256
#define S_   4                     // K-splits per batch (inter-block parallelism)
#define CH   (K_ / (S_ * TPB))     // 2 double-buffered chunks of 256 sites per block
#define CHUNK_F4 ((TPB * D_) / 4)  // 640 float4 per chunk per array
#define TAIL_F4  (CHUNK_F4 - 2 * TPB)  // 128 tail float4s

// Stage 1: one block per (k-split, batch). Double-buffered async global->LDS
// staging with GLOBAL_LOAD_ASYNC_TO_LDS_B128 (ASYNCcnt), closed-form
// determinant det(J)/lam^2 = (lam - i*s)^9 / lam per site (matrix determinant
// lemma; phi^T Y == 0 by construction), block-level complex-product reduction
// into d_ws partials.
__global__ __launch_bounds__(TPB) void cpn_site_kernel(
    const float* __restrict__ phi,     // (B,K,D)
    const float* __restrict__ a,       // (K,D)
    float* __restrict__ out,           // [2*B*K*D] tildeZ interleaved (+detJ tail)
    float* __restrict__ partial)       // [B*S_] complex partial products
{
  __shared__ float sP[2][TPB * D_];    // phi chunk, double buffered
  __shared__ float sA[2][TPB * D_];    // a   chunk, double buffered
  __shared__ float redR[TPB], redI[TPB];

  const int s = blockIdx.x;            // k-split
  const int b = blockIdx.y;            // batch
  const int t = threadIdx.x;

  const int k0 = s * (CH * TPB);       // first site of this block's K-range
  const float* gphi = phi + ((size_t)b * K_ + (size_t)k0) * D_;
  const float* ga   = a   + (size_t)k0 * D_;

  // Generic LDS pointers carry the LDS byte offset in addr[31:0]
  const unsigned ldsPbase = (unsigned)(uintptr_t)&sP[0][0];
  const unsigned ldsAbase = (unsigned)(uintptr_t)&sA[0][0];
  const unsigned bufStride = (unsigned)(TPB * D_ * sizeof(float));

  // Cooperative staging: lane-contiguous float4 -> 512B contiguous per issue.
  auto stage = [&](int j, int buf) {
    const float* pb = gphi + (size_t)j * (TPB * D_);
    const float* ab = ga   + (size_t)j * (TPB * D_);
    const unsigned lp = ldsPbase + (unsigned)buf * bufStride;
    const unsigned la = ldsAbase + (unsigned)buf * bufStride;
#pragma unroll
    for (int i = 0; i < 2; ++i) {
      unsigned voff = 16u * (unsigned)(i * TPB + t);
      asm volatile("global_load_async_to_lds_b128 %0, %1, %2"
                   :: "v"(lp + voff), "v"(voff), "s"(pb) : "memory");
      asm volatile("global_load_async_to_lds_b128 %0, %1, %2"
                   :: "v"(la + voff), "v"(voff), "s"(ab) : "memory");
    }
    if (t < TAIL_F4) {   // 640 = 2*256 + 128 float4s
      unsigned voff = 16u * (unsigned)(2 * TPB + t);
      asm volatile("global_load_async_to_lds_b128 %0, %1, %2"
                   :: "v"(lp + voff), "v"(voff), "s"(pb) : "memory");
      asm volatile("global_load_async_to_lds_b128 %0, %1, %2"
                   :: "v"(la + voff), "v"(voff), "s"(ab) : "memory");
    }
  };

  float pr = 1.0f, pi = 0.0f;          // running complex product of det/lam^2

  stage(0, 0);

  for (int j = 0; j < CH; ++j) {
    const int buf = j & 1;
    // Wait this chunk's async copies, then one barrier makes LDS visible
    // block-wide and orders last iteration's LDS reads before the opposite
    // buffer is overwritten by the prefetch below.
    asm volatile("s_wait_asynccnt 0x0" ::: "memory");
    __syncthreads();
    if (j + 1 < CH) stage(j + 1, buf ^ 1);

    // ---- per-site math, all operands from LDS ----
    const int base = t * D_;
    float x[D_], av[D_];
    float ixx = 0.0f, iax = 0.0f;
#pragma unroll
    for (int d = 0; d < D_; ++d) {
      x[d]  = sP[buf][base + d];
      av[d] = sA[buf][base + d];
      ixx += x[d] * x[d];
      iax += av[d] * x[d];
    }
    const float sc = iax / ixx;

    float y[D_];
    float iyy = 0.0f;
#pragma unroll
    for (int d = 0; d < D_; ++d) {
      y[d] = av[d] - sc * x[d];
      iyy += y[d] * y[d];
    }
    const float lam = sqrtf(1.0f + iyy);

    // tildeZ = lam*phi + i*Y  -> interleaved complex, 5x float4 (16B aligned)
    const int k = k0 + j * TPB + t;
    float4* oz = (float4*)(out + 2ull * ((size_t)(b * K_ + k) * D_));
#pragma unroll
    for (int d = 0; d < D_; d += 2) {
      float4 v;
      v.x = lam * x[d];     v.y = y[d];
      v.z = lam * x[d + 1]; v.w = y[d + 1];
      oz[d >> 1] = v;
    }

    // det(J)/lam^2 = (lam - i*sc)^9 / lam
    const float cr = lam, ci = -sc;
    const float c2r = cr * cr - ci * ci,     c2i = 2.0f * cr * ci;
    const float c4r = c2r * c2r - c2i * c2i, c4i = 2.0f * c2r * c2i;
    const float c8r = c4r * c4r - c4i * c4i, c8i = 2.0f * c4r * c4i;
    const float c9r = c8r * cr - c8i * ci,   c9i = c8r * ci + c8i * cr;
    const float inv = 1.0f / lam;
    const float zr = c9r * inv, zi = c9i * inv;

    const float nr = pr * zr - pi * zi;
    const float ni = pr * zi + pi * zr;
    pr = nr; pi = ni;
  }

  // ---- block-wide complex product reduction -> partial[b*S_ + s] ----
  redR[t] = pr;
  redI[t] = pi;
  __syncthreads();
#pragma unroll
  for (int off = TPB / 2; off > 0; off >>= 1) {
    if (t < off) {
      const float ar = redR[t],       ai = redI[t];
      const float br = redR[t + off], bi = redI[t + off];
      redR[t] = ar * br - ai * bi;
      redI[t] = ar * bi + ai * br;
    }
    __syncthreads();
  }
  if (t == 0) {
    partial[2 * (b * S_ + s)]     = redR[0];
    partial[2 * (b * S_ + s) + 1] = redI[0];
  }
}

// Stage 2: multiply the S_ partial products per batch into detJ[b].
__global__ __launch_bounds__(B_) void cpn_combine_kernel(
    const float* __restrict__ partial, float* __restrict__ out)
{
  const int b = threadIdx.x;
  if (b >= B_) return;
  float pr = 1.0f, pi = 0.0f;
#pragma unroll
  for (int s = 0; s < S_; ++s) {
    const float zr = partial[2 * (b * S_ + s)];
    const float zi = partial[2 * (b * S_ + s) + 1];
    const float nr = pr * zr - pi * zi;
    const float ni = pr * zi + pi * zr;
    pr = nr; pi = ni;
  }
  const size_t off = 2ull * B_ * K_ * D_;
  out[off + 2 * b]     = pr;
  out[off + 2 * b + 1] = pi;
}

extern "C" void kernel_launch(void* const* d_in, const int* in_sizes, int n_in,
                              void* d_out, int out_size, void* d_ws, size_t ws_size,
                              hipStream_t stream) {
  (void)in_sizes; (void)n_in; (void)out_size; (void)ws_size;  // ws use: 4 KB
  const float* phi = (const float*)d_in[0];   // (B,K,D,1) float32
  const float* a   = (const float*)d_in[1];   // (K,D,1)  float32
  float* out = (float*)d_out;                 // tildeZ complex interleaved, then detJ
  float* ws  = (float*)d_ws;                  // B_*S_ complex partials (written then read)

  cpn_site_kernel<<<dim3(S_, B_), dim3(TPB), 0, stream>>>(phi, a, out, ws);
  cpn_combine_kernel<<<dim3(1), dim3(B_), 0, stream>>>(ws, out);
}